// Involution2d_60009283059831
// MI455X (gfx1250) — compile-verified
//
#include <hip/hip_runtime.h>
#include <math.h>

// ---------------- problem constants ----------------
#define KSZ   7
#define PAD   3
#define GRP   12
#define CIN   192
#define CG    16          // CIN/GRP
#define CR    48          // reduce channels
#define SPAN  588         // KSZ*KSZ*GRP
#define BATCH 16
#define HH    32
#define WW    32
#define HW    1024
#define FLAT  (CIN*HW)    // 196608
#define F1    512
#define F2    128
#define NCLS  17
#define EPSV  1e-5f

// FC1 K-split for parallelism (32 j-tiles alone cannot saturate 23.3 TB/s)
#define NCHUNK 96
#define KCHUNK (FLAT/NCHUNK)   // 2048

typedef float v2f __attribute__((ext_vector_type(2)));
typedef float v8f __attribute__((ext_vector_type(8)));

// ---------------- workspace layout (float offsets) ----------------
#define OFF_RED 0u                 // 786432  : red  [B][CR][HW]
#define OFF_AS  786432u            // 48      : BN scale for red
#define OFF_BS  786496u            // 48      : BN shift for red
#define OFF_KER 786560u            // 9633792 : ker  [B][SPAN][HW]
#define OFF_INV 10420352u          // 3145728 : involution out [B][CIN][HW]
#define OFF_P   13566080u          // 786432  : FC1 partials [NCHUNK][F1*BATCH]
#define OFF_Y1  14352512u          // 8192    : y1 [F1][BATCH]
#define OFF_A1  14360704u          // 512
#define OFF_B1N 14361216u          // 512
#define OFF_H1  14361728u          // 8192    : h1 [BATCH][F1]
#define OFF_Y2  14369920u          // 2048    : y2 [BATCH][F2]

// ---------------- kernel 1: red = Wr @ x (per pixel 1x1 conv) ----------------
// grid 64 (b*4 + pixel-tile), block 256 threads = 256 pixels, 48 accumulators
__global__ void k_reduce(const float* __restrict__ x, const float* __restrict__ Wr,
                         float* __restrict__ red) {
  int b  = blockIdx.x >> 2;
  int p  = ((blockIdx.x & 3) << 8) + threadIdx.x;
  float acc[CR];
#pragma unroll
  for (int o = 0; o < CR; ++o) acc[o] = 0.f;
  const float* xb = x + (size_t)b * CIN * HW + p;
  for (int c = 0; c < CIN; ++c) {
    float xv = xb[(size_t)c * HW];
#pragma unroll
    for (int o = 0; o < CR; ++o) acc[o] += Wr[o * CIN + c] * xv;
  }
  float* rb = red + (size_t)b * CR * HW + p;
#pragma unroll
  for (int o = 0; o < CR; ++o) rb[(size_t)o * HW] = acc[o];
}

// ---------------- kernel 2: per-channel BN stats of red ----------------
// grid 48 (one channel), block 256
__global__ void k_bnstat(const float* __restrict__ red, const float* __restrict__ gamma,
                         const float* __restrict__ beta,
                         float* __restrict__ as_, float* __restrict__ bs_) {
  __shared__ float ssum[256], ssq[256];
  int o = blockIdx.x;
  float s = 0.f, s2 = 0.f;
  for (int i = threadIdx.x; i < BATCH * HW; i += 256) {
    int b = i >> 10, hw = i & 1023;
    float v = red[((size_t)b * CR + o) * HW + hw];
    s += v; s2 += v * v;
  }
  ssum[threadIdx.x] = s; ssq[threadIdx.x] = s2;
  __syncthreads();
  for (int st = 128; st > 0; st >>= 1) {
    if (threadIdx.x < st) {
      ssum[threadIdx.x] += ssum[threadIdx.x + st];
      ssq[threadIdx.x]  += ssq[threadIdx.x + st];
    }
    __syncthreads();
  }
  if (threadIdx.x == 0) {
    float inv_n = 1.f / (float)(BATCH * HW);
    float mean  = ssum[0] * inv_n;
    float var   = ssq[0] * inv_n - mean * mean;
    float a     = gamma[o] / sqrtf(var + EPSV);
    as_[o] = a;
    bs_[o] = beta[o] - mean * a;
  }
}

// ---------------- kernel 3: ker = Ws @ relu(bn(red)) ----------------
// grid (64, 28): x = b*4+ptile, y = s-tile of 21; block 256 = pixels
#define STILE 21
__global__ void k_span(const float* __restrict__ red, const float* __restrict__ Ws,
                       const float* __restrict__ as_, const float* __restrict__ bs_,
                       float* __restrict__ ker) {
  __shared__ float rn[CR * 256];    // 48 KB normalized activations
  int b  = blockIdx.x >> 2;
  int p0 = (blockIdx.x & 3) << 8;
  int s0 = blockIdx.y * STILE;
  int t  = threadIdx.x;
  for (int o = 0; o < CR; ++o) {
    float v = red[((size_t)b * CR + o) * HW + p0 + t];
    rn[o * 256 + t] = fmaxf(as_[o] * v + bs_[o], 0.f);
  }
  __syncthreads();
  float acc[STILE];
#pragma unroll
  for (int i = 0; i < STILE; ++i) acc[i] = 0.f;
  for (int o = 0; o < CR; ++o) {
    float v = rn[o * 256 + t];
#pragma unroll
    for (int i = 0; i < STILE; ++i) acc[i] += Ws[(s0 + i) * CR + o] * v;
  }
#pragma unroll
  for (int i = 0; i < STILE; ++i)
    ker[((size_t)b * SPAN + s0 + i) * HW + p0 + t] = acc[i];
}

// ---------------- kernel 4: involution apply ----------------
// grid (B*GRP, HH), block 512: cg = tid>>5, w = tid&31
__global__ void k_invol(const float* __restrict__ x, const float* __restrict__ ker,
                        float* __restrict__ out) {
  __shared__ float ks[49 * 32];
  int bg = blockIdx.x;
  int b = bg / GRP, g = bg % GRP;
  int h = blockIdx.y;
  int tid = threadIdx.x;
  for (int i = tid; i < 49 * 32; i += 512) {
    int kk = i >> 5, ww = i & 31;
    ks[i] = ker[((size_t)b * SPAN + g * 49 + kk) * HW + h * WW + ww];
  }
  __syncthreads();
  int cg = tid >> 5, w = tid & 31;
  int c = g * CG + cg;
  const float* xb = x + ((size_t)b * CIN + c) * HW;
  float acc = 0.f;
#pragma unroll
  for (int kh = 0; kh < KSZ; ++kh) {
    int h2 = h + kh - PAD;
    if (h2 < 0 || h2 >= HH) continue;
#pragma unroll
    for (int kw = 0; kw < KSZ; ++kw) {
      int w2 = w + kw - PAD;
      if (w2 < 0 || w2 >= WW) continue;
      acc += ks[(kh * KSZ + kw) * 32 + w] * xb[h2 * WW + w2];
    }
  }
  out[((size_t)b * CIN + c) * HW + h * WW + w] = acc;
}

// ---------------- kernel 5: FC1 via fp32 WMMA, K-split partials ----------------
// grid (32 j-tiles, NCHUNK k-chunks), block 32 (one wave32)
// D(16x16) = A(16x4 of W1 rows) x B(4x16 of flat^T) + C, accumulated over KCHUNK
__global__ void k_fc1_wmma(const float* __restrict__ W1, const float* __restrict__ flat,
                           float* __restrict__ part) {
  int j0   = blockIdx.x << 4;
  int kc   = blockIdx.y;
  int lane = threadIdx.x;            // 0..31
  int m    = lane & 15;              // A: row M / B: col N (batch)
  int kb   = (lane >> 4) << 1;       // K half select: {0,2}
  const float* arow = W1   + (size_t)(j0 + m) * FLAT + kb;   // W1[j0+m][k+kb..]
  const float* brow = flat + (size_t)m * FLAT + kb;          // flat[b=m][k+kb..]
  v8f c = {};
  int k0 = kc * KCHUNK;
  for (int k = k0; k < k0 + KCHUNK; k += 4) {
    v2f a  = *(const v2f*)(arow + k);
    v2f bf = *(const v2f*)(brow + k);
    c = __builtin_amdgcn_wmma_f32_16x16x4_f32(false, a, false, bf, (short)0, c,
                                              false, false);
  }
  // D layout: VGPR v, lanes 0-15 -> M=v, N=lane; lanes 16-31 -> M=v+8, N=lane-16
  float* pb = part + (size_t)kc * (F1 * BATCH);
  int hi = lane >> 4;
#pragma unroll
  for (int v = 0; v < 8; ++v) {
    int mm = v + (hi << 3);
    pb[(j0 + mm) * BATCH + m] = c[v];
  }
}

// ---------------- kernel 6: reduce FC1 partials + bias ----------------
__global__ void k_fc1_reduce(const float* __restrict__ part, const float* __restrict__ b1,
                             float* __restrict__ y1) {
  int gid = blockIdx.x * 256 + threadIdx.x;   // 8192 = F1*BATCH
  int j = gid >> 4;
  float s = b1[j];
  for (int kc = 0; kc < NCHUNK; ++kc) s += part[(size_t)kc * (F1 * BATCH) + gid];
  y1[gid] = s;
}

// ---------------- kernel 7: BN stats over batch for FC1 ----------------
__global__ void k_fc1_stat(const float* __restrict__ y1, const float* __restrict__ g1,
                           const float* __restrict__ be1,
                           float* __restrict__ a1, float* __restrict__ bb1) {
  int j = blockIdx.x * 256 + threadIdx.x;
  if (j >= F1) return;
  float s = 0.f, s2 = 0.f;
#pragma unroll
  for (int b = 0; b < BATCH; ++b) { float v = y1[j * 16 + b]; s += v; s2 += v * v; }
  float mean = s * (1.f / BATCH);
  float var  = s2 * (1.f / BATCH) - mean * mean;
  float a = g1[j] / sqrtf(var + EPSV);
  a1[j] = a; bb1[j] = be1[j] - mean * a;
}

// ---------------- kernel 8: apply BN1 + ReLU, transpose to [B][F1] ----------------
__global__ void k_fc1_apply(const float* __restrict__ y1, const float* __restrict__ a1,
                            const float* __restrict__ bb1, float* __restrict__ h1) {
  int gid = blockIdx.x * 256 + threadIdx.x;   // 8192
  int b = gid >> 9, j = gid & 511;
  h1[gid] = fmaxf(a1[j] * y1[j * 16 + b] + bb1[j], 0.f);
}

// ---------------- kernel 9: FC2 (tiny: 16x128, K=512) ----------------
__global__ void k_fc2(const float* __restrict__ h1, const float* __restrict__ W2,
                      const float* __restrict__ b2, float* __restrict__ y2) {
  int gid = blockIdx.x * 256 + threadIdx.x;   // 2048
  int b = gid >> 7, i = gid & 127;
  float acc = b2[i];
  const float* hr = h1 + b * F1;
  const float* wr = W2 + i * F1;
  for (int j = 0; j < F1; ++j) acc += hr[j] * wr[j];
  y2[gid] = acc;
}

// ---------------- kernel 10: BN2 stats + ReLU + FC3 head ----------------
__global__ void k_head(const float* __restrict__ y2, const float* __restrict__ g2,
                       const float* __restrict__ be2, const float* __restrict__ W3,
                       const float* __restrict__ b3, float* __restrict__ out) {
  __shared__ float a2s[F2], b2s[F2];
  int tid = threadIdx.x;
  if (tid < F2) {
    float s = 0.f, s2 = 0.f;
#pragma unroll
    for (int b = 0; b < BATCH; ++b) { float v = y2[b * F2 + tid]; s += v; s2 += v * v; }
    float mean = s * (1.f / BATCH);
    float var  = s2 * (1.f / BATCH) - mean * mean;
    float a = g2[tid] / sqrtf(var + EPSV);
    a2s[tid] = a; b2s[tid] = be2[tid] - mean * a;
  }
  __syncthreads();
  if (tid < BATCH * NCLS) {
    int b = tid / NCLS, t = tid % NCLS;
    float acc = b3[t];
    for (int i = 0; i < F2; ++i) {
      float h = fmaxf(a2s[i] * y2[b * F2 + i] + b2s[i], 0.f);
      acc += h * W3[t * F2 + i];
    }
    out[tid] = acc;
  }
}

// ---------------- launcher ----------------
extern "C" void kernel_launch(void* const* d_in, const int* in_sizes, int n_in,
                              void* d_out, int out_size, void* d_ws, size_t ws_size,
                              hipStream_t stream) {
  const float* x       = (const float*)d_in[0];
  const float* Wr      = (const float*)d_in[1];
  const float* gamma_s = (const float*)d_in[2];
  const float* beta_s  = (const float*)d_in[3];
  const float* Ws      = (const float*)d_in[4];
  const float* W1      = (const float*)d_in[5];
  const float* b1      = (const float*)d_in[6];
  const float* g1      = (const float*)d_in[7];
  const float* beta1   = (const float*)d_in[8];
  const float* W2      = (const float*)d_in[9];
  const float* b2      = (const float*)d_in[10];
  const float* g2      = (const float*)d_in[11];
  const float* beta2   = (const float*)d_in[12];
  const float* W3      = (const float*)d_in[13];
  const float* b3      = (const float*)d_in[14];
  float* out = (float*)d_out;
  float* w   = (float*)d_ws;

  float* red  = w + OFF_RED;
  float* as_  = w + OFF_AS;
  float* bs_  = w + OFF_BS;
  float* ker  = w + OFF_KER;
  float* inv  = w + OFF_INV;
  float* part = w + OFF_P;
  float* y1   = w + OFF_Y1;
  float* a1   = w + OFF_A1;
  float* bb1  = w + OFF_B1N;
  float* h1   = w + OFF_H1;
  float* y2   = w + OFF_Y2;

  k_reduce    <<<64,               256, 0, stream>>>(x, Wr, red);
  k_bnstat    <<<48,               256, 0, stream>>>(red, gamma_s, beta_s, as_, bs_);
  k_span      <<<dim3(64, 28),     256, 0, stream>>>(red, Ws, as_, bs_, ker);
  k_invol     <<<dim3(BATCH*GRP, HH), 512, 0, stream>>>(x, ker, inv);
  k_fc1_wmma  <<<dim3(F1/16, NCHUNK), 32, 0, stream>>>(W1, inv, part);
  k_fc1_reduce<<<32,               256, 0, stream>>>(part, b1, y1);
  k_fc1_stat  <<<2,                256, 0, stream>>>(y1, g1, beta1, a1, bb1);
  k_fc1_apply <<<32,               256, 0, stream>>>(y1, a1, bb1, h1);
  k_fc2       <<<8,                256, 0, stream>>>(h1, W2, b2, y2);
  k_head      <<<1,                512, 0, stream>>>(y2, g2, beta2, W3, b3, out);
}